// MSCSA_58119497450369
// MI455X (gfx1250) — compile-verified
//
#include <hip/hip_runtime.h>
#include <hip/hip_bf16.h>

typedef __attribute__((ext_vector_type(2))) float v2f;
typedef __attribute__((ext_vector_type(8))) float v8f;

#define C_   64
#define H_   224
#define W_   224
#define HW_  (224 * 224)
#define XSTRIDE 244     // 224 line + 2*10 zero pad
#define PADP 10
#define LINES_PER_WG 4

// -------------------------------------------------------------------------
// Stage 0: zero the workspace accumulators (ws is poisoned by the harness)
// -------------------------------------------------------------------------
__global__ void mscsa_zero(float* __restrict__ p, int n) {
    int i = blockIdx.x * 256 + threadIdx.x;
    if (i < n) p[i] = 0.f;
}

// -------------------------------------------------------------------------
// Stage 1: fused depthwise conv + Gram (q·k^T via V_WMMA_F32_16X16X4_F32)
//          + v column-sums, for TWO branches (lrb + srb) of one direction.
//  grid = (224/LINES_PER_WG, 8 batches), block = 256 threads (8 wave32)
//  wave w: branch = w/4, Gram tile-row = w%4; each wave accumulates 4
//  16x16 f32 tiles -> full 64x64 Gram per branch.
// -------------------------------------------------------------------------
__global__ __launch_bounds__(256)
void mscsa_stage1(const float* __restrict__ x,
                  const float* __restrict__ wqA, const float* __restrict__ wkA,
                  const float* __restrict__ wvA,
                  const float* __restrict__ wqB, const float* __restrict__ wkB,
                  const float* __restrict__ wvB,
                  float* __restrict__ gramA, float* __restrict__ gramB,
                  float* __restrict__ vsA,   float* __restrict__ vsB,
                  int pos_stride, int line_stride)
{
    extern __shared__ float smem[];
    float* xs = smem;             // [64][244] padded line cache
    float* qs = smem + 15616;     // [2 branches][64 ch][4 pos] staging (A)
    float* ks = smem + 16128;     // [2 branches][64 ch][4 pos] staging (B)

    const int t    = threadIdx.x;
    const int lane = t & 31;
    const int wv   = t >> 5;
    const int br   = wv >> 2;     // 0 = branch A (lrb), 1 = branch B (srb)
    const int ti   = wv & 3;      // Gram tile row
    const int sc   = t >> 2;      // staging channel 0..63
    const int sr   = t & 3;       // staging position-in-chunk 0..3
    const int b    = blockIdx.y;
    const long xbase = (long)b * C_ * HW_;

    v8f acc[4] = {};              // 4 accumulated 16x16 tiles for this wave
    float vaccA = 0.f, vaccB = 0.f;

    for (int l0 = 0; l0 < LINES_PER_WG; ++l0) {
        const int line  = blockIdx.x * LINES_PER_WG + l0;
        const long lbase = xbase + (long)line * line_stride;

        // prefetch next line (emits global_prefetch_b8)
        if (l0 + 1 < LINES_PER_WG)
            __builtin_prefetch(&x[lbase + line_stride + (long)(t & 63) * HW_], 0, 1);

        // cooperative zero-padded line load: xs[c][PADP + pos]
        for (int i = t; i < C_ * XSTRIDE; i += 256) {
            const int c = i / XSTRIDE, p = i % XSTRIDE;
            float v = 0.f;
            if (p >= PADP && p < PADP + H_)
                v = x[lbase + (long)c * HW_ + (long)(p - PADP) * pos_stride];
            xs[i] = v;
        }
        __syncthreads();

        for (int kc = 0; kc < 56; ++kc) {
            const int row = kc * 4 + sr;
            const float* xc = &xs[sc * XSTRIDE + row];

            // depthwise convs (cross-correlation, 'same' padding)
            float qA = 0.f, qB = 0.f, kA = 0.f, kB = 0.f;
#pragma unroll
            for (int i = 0; i < 7; ++i) {           // q: 7 taps, pad 3
                const float xv = xc[i + 7];
                qA += wqA[sc * 7 + i] * xv;
                qB += wqB[sc * 7 + i] * xv;
            }
#pragma unroll
            for (int i = 0; i < 11; ++i) {          // k: 11 taps, pad 5
                const float xv = xc[i + 5];
                kA += wkA[sc * 11 + i] * xv;
                kB += wkB[sc * 11 + i] * xv;
            }
#pragma unroll
            for (int i = 0; i < 21; ++i) {          // v: 21 taps, pad 10
                const float xv = xc[i];
                vaccA += wvA[sc * 21 + i] * xv;
                vaccB += wvB[sc * 21 + i] * xv;
            }
            qs[sc * 4 + sr]       = qA;
            qs[256 + sc * 4 + sr] = qB;
            ks[sc * 4 + sr]       = kA;
            ks[256 + sc * 4 + sr] = kB;
            __syncthreads();

            // WMMA rank-4 update of the 64x64 Gram (f32, full precision)
            const float* qp = qs + br * 256;
            const float* kp = ks + br * 256;
            const int rsel = lane & 15;
            const int k0   = (lane >> 4) << 1;      // 16x4 f32 A layout
            v2f a;
            a.x = qp[(ti * 16 + rsel) * 4 + k0];
            a.y = qp[(ti * 16 + rsel) * 4 + k0 + 1];
#pragma unroll
            for (int j = 0; j < 4; ++j) {
                v2f bb;
                bb.x = kp[(j * 16 + rsel) * 4 + k0];
                bb.y = kp[(j * 16 + rsel) * 4 + k0 + 1];
                acc[j] = __builtin_amdgcn_wmma_f32_16x16x4_f32(
                    false, a, false, bb, (short)0, acc[j], false, false);
            }
            __syncthreads();
        }
    }

    // v-sum reduction over the 4 position-phases, then global accumulate
    qs[sc * 4 + sr] = vaccA;
    ks[sc * 4 + sr] = vaccB;
    __syncthreads();
    if (t < C_) {
        const float sA = qs[t * 4] + qs[t * 4 + 1] + qs[t * 4 + 2] + qs[t * 4 + 3];
        const float sB = ks[t * 4] + ks[t * 4 + 1] + ks[t * 4 + 2] + ks[t * 4 + 3];
        atomicAdd(&vsA[b * C_ + t], sA);
        atomicAdd(&vsB[b * C_ + t], sB);
    }

    // Gram accumulate: C/D layout lane%16 = N, lane/16 selects M block of 8
    float* g = (br ? gramB : gramA) + (long)b * 4096;
#pragma unroll
    for (int j = 0; j < 4; ++j) {
#pragma unroll
        for (int r = 0; r < 8; ++r) {
            const int m = ((lane >> 4) << 3) + r;
            const int n = (lane & 15) + j * 16;
            atomicAdd(&g[(ti * 16 + m) * 64 + n], acc[j][r]);
        }
    }
}

// -------------------------------------------------------------------------
// Stage 2: per-(b,c) softmax over Gram row (full row for lrb heads=1;
//          8-wide diagonal block for srb heads=8), dot with v-mean,
//          mix_w combine, sigmoid. grid = 8, block = 64.
// -------------------------------------------------------------------------
__global__ void mscsa_stage2(const float* __restrict__ gram,
                             const float* __restrict__ vsum,
                             const float* __restrict__ mixw,
                             float* __restrict__ sout)
{
    const int b = blockIdx.x;
    const int c = threadIdx.x;
    float mixv = 0.f;
#pragma unroll
    for (int cb = 0; cb < 4; ++cb) {
        const float* g  = gram + ((long)(cb * 8 + b)) * 4096 + c * 64;
        const float* vm = vsum + (cb * 8 + b) * 64;
        const float scale = (cb < 2) ? 0.125f : 0.35355339059327373f; // hd^-0.5
        const int e0 = (cb < 2) ? 0 : (c & ~7);
        const int ne = (cb < 2) ? 64 : 8;
        float mx = -3.4e38f;
        for (int e = 0; e < ne; ++e) mx = fmaxf(mx, g[e0 + e] * scale);
        float se = 0.f, sv = 0.f;
        for (int e = 0; e < ne; ++e) {
            const float ex = __expf(g[e0 + e] * scale - mx);
            se += ex;
            sv += ex * (vm[e0 + e] * (1.f / 50176.f));
        }
        mixv += mixw[cb] * (sv / se);
    }
    sout[b * 64 + c] = 1.f / (1.f + __expf(-mixv));
}

// -------------------------------------------------------------------------
// Stage 3: out = sigmoid_gate[plane] * x, float4 vectorized.
// 12544 float4 per (b,c) plane == 49 blocks of 256 -> plane-uniform blocks.
// -------------------------------------------------------------------------
__global__ __launch_bounds__(256)
void mscsa_stage3(const float4* __restrict__ x4, const float* __restrict__ s,
                  float4* __restrict__ o4)
{
    const long i = (long)blockIdx.x * 256 + threadIdx.x;
    const int plane = (int)(i / 12544);
    const float sc = s[plane];
    float4 v = x4[i];
    v.x *= sc; v.y *= sc; v.z *= sc; v.w *= sc;
    o4[i] = v;
}

// -------------------------------------------------------------------------
extern "C" void kernel_launch(void* const* d_in, const int* in_sizes, int n_in,
                              void* d_out, int out_size, void* d_ws, size_t ws_size,
                              hipStream_t stream)
{
    (void)in_sizes; (void)n_in; (void)out_size; (void)ws_size;
    const float* x    = (const float*)d_in[0];
    const float* mixw = (const float*)d_in[13];

    float* ws   = (float*)d_ws;
    float* gram = ws;               // [4 combos][8][64][64]  = 131072 f
    float* vsum = ws + 131072;      // [4 combos][8][64]      =   2048 f
    float* sarr = ws + 133120;      // [8][64]                =    512 f

    mscsa_zero<<<522, 256, 0, stream>>>(ws, 133632);

    const size_t shbytes = 16640 * sizeof(float);   // 66.5 KB dynamic LDS

    // vertical direction -> combos 0 (lrb_h) and 2 (srb_h)
    mscsa_stage1<<<dim3(224 / LINES_PER_WG, 8), 256, shbytes, stream>>>(
        x,
        (const float*)d_in[1], (const float*)d_in[3], (const float*)d_in[5],
        (const float*)d_in[7], (const float*)d_in[9], (const float*)d_in[11],
        gram + 0 * 32768, gram + 2 * 32768, vsum + 0 * 512, vsum + 2 * 512,
        /*pos_stride=*/W_, /*line_stride=*/1);

    // horizontal direction -> combos 1 (lrb_w) and 3 (srb_w)
    mscsa_stage1<<<dim3(224 / LINES_PER_WG, 8), 256, shbytes, stream>>>(
        x,
        (const float*)d_in[2], (const float*)d_in[4], (const float*)d_in[6],
        (const float*)d_in[8], (const float*)d_in[10], (const float*)d_in[12],
        gram + 1 * 32768, gram + 3 * 32768, vsum + 1 * 512, vsum + 3 * 512,
        /*pos_stride=*/1, /*line_stride=*/W_);

    mscsa_stage2<<<8, 64, 0, stream>>>(gram, vsum, mixw, sarr);

    mscsa_stage3<<<25088, 256, 0, stream>>>((const float4*)x, sarr, (float4*)d_out);
}